// MinimalSSM_48722109006071
// MI455X (gfx1250) — compile-verified
//
#include <hip/hip_runtime.h>

// MinimalSSM on gfx1250: chunked parallel scan + fp32 WMMA GEMMs.
// T=262144, D_IN=64, D_STATE=128, dt=0.01
#define T_LEN   262144
#define D_INP   64
#define D_ST    128
#define DT_C    0.01f
#define CHUNK   256                 // time steps per block
#define SUBT    64                  // sub-tile rows held in LDS
#define NSUB    (CHUNK / SUBT)      // 4
#define NCHUNK  (T_LEN / CHUNK)     // 1024
#define GSIZE   32                  // chunks per scan group
#define NGROUP  (NCHUNK / GSIZE)    // 32
#define LSTR    66                  // padded LDS row stride (floats) -> conflict-free

typedef __attribute__((ext_vector_type(2))) float v2f;
typedef __attribute__((ext_vector_type(8))) float v8f;

__device__ __forceinline__ v8f wmma_f32(v2f a, v2f b, v8f c) {
  // V_WMMA_F32_16X16X4_F32 : D = A(16x4) * B(4x16) + C(16x16), all fp32
  return __builtin_amdgcn_wmma_f32_16x16x4_f32(
      /*neg_a=*/false, a, /*neg_b=*/false, b,
      /*c_mod=*/(short)0, c, /*reuse_a=*/false, /*reuse_b=*/false);
}

// Compute Bu = U @ B^T for 64 time rows starting at global row t0.
// Result written transposed into ldsT[state][LSTR] (time-minor, padded).
// Block = 128 threads = 4 waves; wave w owns time-row-tile w.
__device__ __forceinline__ void bu_subtile(const float* __restrict__ U,
                                           const float* __restrict__ Bm,
                                           int t0, float* ldsT) {
  const int wave = threadIdx.x >> 5;
  const int lane = threadIdx.x & 31;
  const int half = lane >> 4;       // which K-pair group
  const int l15  = lane & 15;
  const int rowA = t0 + wave * 16 + l15;     // global time row for A fragment

  v2f a[16];                                  // A frags for full K=64
#pragma unroll
  for (int ks = 0; ks < 16; ++ks) {
    const int k = ks * 4 + half * 2;
    a[ks] = *(const v2f*)(U + (size_t)rowA * D_INP + k);
  }
#pragma unroll
  for (int nt = 0; nt < 8; ++nt) {            // 8 state tiles of 16
    const int sidx = nt * 16 + l15;           // state column (B^T col = B row)
    v8f acc = {};
#pragma unroll
    for (int ks = 0; ks < 16; ++ks) {
      const int k = ks * 4 + half * 2;
      v2f b = *(const v2f*)(Bm + (size_t)sidx * D_INP + k);  // B[s][k],B[s][k+1]
      acc = wmma_f32(a[ks], b, acc);
    }
#pragma unroll
    for (int i = 0; i < 8; ++i) {             // D layout: VGPR i -> M = i + 8*half
      const int r = wave * 16 + i + half * 8; // time row within sub-tile
      ldsT[sidx * LSTR + r] = acc[i];
    }
  }
}

// Phase 1: per-chunk local scan (h starts at 0), emit chunk sums.
__global__ void __launch_bounds__(128)
ssm_phase1(const float* __restrict__ U, const float* __restrict__ Alog,
           const float* __restrict__ Bm, float* __restrict__ Ssum) {
  __shared__ float ldsT[D_ST * LSTR];
  const int chunk = blockIdx.x;
  const int s = threadIdx.x;                  // one state per thread
  const float a = expf(Alog[s] * DT_C);
  float h = 0.f;
  for (int sub = 0; sub < NSUB; ++sub) {
    bu_subtile(U, Bm, chunk * CHUNK + sub * SUBT, ldsT);
    __syncthreads();
#pragma unroll 8
    for (int t = 0; t < SUBT; ++t)
      h = fmaf(a, h, ldsT[s * LSTR + t]);
    __syncthreads();
  }
  Ssum[(size_t)chunk * D_ST + s] = h;
}

// Phase 2a: scan 32 chunk-sums per group, register-resident (bulk loads).
__global__ void __launch_bounds__(128)
ssm_phase2a(const float* __restrict__ Alog, const float* __restrict__ Ssum,
            float* __restrict__ Lc, float* __restrict__ Gs) {
  const int b = blockIdx.x;                   // group
  const int s = threadIdx.x;
  const float aL = expf(Alog[s] * (DT_C * (float)CHUNK));
  float v[GSIZE];
#pragma unroll
  for (int j = 0; j < GSIZE; ++j)
    v[j] = Ssum[(size_t)(b * GSIZE + j) * D_ST + s];
  float l = 0.f;
#pragma unroll
  for (int j = 0; j < GSIZE; ++j) {
    Lc[(size_t)(b * GSIZE + j) * D_ST + s] = l;   // local prefix (zero-init)
    l = fmaf(aL, l, v[j]);
  }
  Gs[(size_t)b * D_ST + s] = l;               // group sum
}

// Phase 2b: scan 32 group sums (with h0), emit group carries + h_final.
__global__ void __launch_bounds__(128)
ssm_phase2b(const float* __restrict__ Alog, const float* __restrict__ h0,
            const float* __restrict__ Gs, float* __restrict__ Cs,
            float* __restrict__ hfin) {
  const int s = threadIdx.x;
  const float aG = expf(Alog[s] * (DT_C * (float)(CHUNK * GSIZE)));
  float g[NGROUP];
#pragma unroll
  for (int j = 0; j < NGROUP; ++j) g[j] = Gs[(size_t)j * D_ST + s];
  float c = h0[s];
#pragma unroll
  for (int j = 0; j < NGROUP; ++j) {
    Cs[(size_t)j * D_ST + s] = c;             // carry entering group j
    c = fmaf(aG, c, g[j]);
  }
  hfin[s] = c;                                 // final hidden state
}

// Phase 3: recompute Bu, apply true carry, scan -> H in LDS,
// then Y = H @ C^T + U @ D^T via WMMA.
__global__ void __launch_bounds__(128)
ssm_phase3(const float* __restrict__ U, const float* __restrict__ Alog,
           const float* __restrict__ Bm, const float* __restrict__ Cm,
           const float* __restrict__ Dm, const float* __restrict__ Lc,
           const float* __restrict__ Cs, float* __restrict__ Y) {
  __shared__ float ldsT[D_ST * LSTR];
  const int chunk = blockIdx.x;
  const int s = threadIdx.x;
  const int grp = chunk >> 5;                  // chunk / GSIZE
  const int jin = chunk & (GSIZE - 1);
  const float a  = expf(Alog[s] * DT_C);
  const float aJ = expf(Alog[s] * (DT_C * (float)CHUNK * (float)jin)); // aL^jin
  // true carry entering this chunk = local prefix + aL^jin * group carry
  float h = fmaf(aJ, Cs[(size_t)grp * D_ST + s], Lc[(size_t)chunk * D_ST + s]);

  const int wave = threadIdx.x >> 5;
  const int lane = threadIdx.x & 31;
  const int half = lane >> 4;
  const int l15  = lane & 15;

  for (int sub = 0; sub < NSUB; ++sub) {
    const int t0 = chunk * CHUNK + sub * SUBT;
    bu_subtile(U, Bm, t0, ldsT);
    __syncthreads();
    // sequential scan over this sub-tile; write H back in place
#pragma unroll 8
    for (int t = 0; t < SUBT; ++t) {
      h = fmaf(a, h, ldsT[s * LSTR + t]);
      ldsT[s * LSTR + t] = h;
    }
    __syncthreads();
    // Y(64x64) = H(64x128) @ C^T + U(64x64) @ D^T, wave = time row tile
    const int rloc = wave * 16 + l15;          // time row within sub-tile
#pragma unroll
    for (int nt = 0; nt < 4; ++nt) {           // 4 output-column tiles
      const int j = nt * 16 + l15;             // output column (d_in index)
      v8f acc = {};
#pragma unroll
      for (int ks = 0; ks < 32; ++ks) {        // K = 128 over states
        const int k = ks * 4 + half * 2;
        v2f ha;
        ha.x = ldsT[k * LSTR + rloc];          // H[t][k]   (transposed LDS)
        ha.y = ldsT[(k + 1) * LSTR + rloc];    // H[t][k+1]
        v2f cb = *(const v2f*)(Cm + (size_t)j * D_ST + k);  // C[j][k..k+1]
        acc = wmma_f32(ha, cb, acc);
      }
#pragma unroll
      for (int ks = 0; ks < 16; ++ks) {        // K = 64 over inputs (Du term)
        const int k = ks * 4 + half * 2;
        v2f ua = *(const v2f*)(U + (size_t)(t0 + rloc) * D_INP + k);
        v2f db = *(const v2f*)(Dm + (size_t)j * D_INP + k);
        acc = wmma_f32(ua, db, acc);
      }
#pragma unroll
      for (int i = 0; i < 8; ++i) {
        const int r = wave * 16 + i + half * 8;
        Y[(size_t)(t0 + r) * D_INP + nt * 16 + l15] = acc[i];
      }
    }
    __syncthreads();
  }
}

extern "C" void kernel_launch(void* const* d_in, const int* in_sizes, int n_in,
                              void* d_out, int out_size, void* d_ws, size_t ws_size,
                              hipStream_t stream) {
  const float* U    = (const float*)d_in[0];   // (T, 64)
  const float* Alog = (const float*)d_in[1];   // (128,)
  const float* Bm   = (const float*)d_in[2];   // (128, 64)
  const float* Cm   = (const float*)d_in[3];   // (64, 128)
  const float* Dm   = (const float*)d_in[4];   // (64, 64)
  const float* h0   = (const float*)d_in[5];   // (128,)
  float* Y    = (float*)d_out;                  // (T,64) then h_final(128)
  float* hfin = Y + (size_t)T_LEN * D_INP;

  float* Ssum = (float*)d_ws;                           // NCHUNK * 128
  float* Lc   = Ssum + (size_t)NCHUNK * D_ST;           // NCHUNK * 128
  float* Gs   = Lc   + (size_t)NCHUNK * D_ST;           // NGROUP * 128
  float* Cs   = Gs   + (size_t)NGROUP * D_ST;           // NGROUP * 128

  ssm_phase1 <<<dim3(NCHUNK), dim3(128), 0, stream>>>(U, Alog, Bm, Ssum);
  ssm_phase2a<<<dim3(NGROUP), dim3(128), 0, stream>>>(Alog, Ssum, Lc, Gs);
  ssm_phase2b<<<dim3(1),      dim3(128), 0, stream>>>(Alog, h0, Gs, Cs, hfin);
  ssm_phase3 <<<dim3(NCHUNK), dim3(128), 0, stream>>>(U, Alog, Bm, Cm, Dm, Lc, Cs, Y);
}